// NormXCorr_78005196030039
// MI455X (gfx1250) — compile-verified
//
#include <hip/hip_runtime.h>
#include <math.h>

// NormXCorr on MI455X (gfx1250), fp32 WMMA (v_wmma_f32_16x16x4_f32).
//   per (b,d,x):  D[ac][y] = sum_k Ypatch[ac][k] * Xpatch[y][k]   (M=120,N=24,K=25)
//   epilogue:     out = (D/25 - muX*muY) * rsX * rsY
// Bound by the 566 MB output store (~24us @ 23.3 TB/s); inner loop is engineered
// so issue time matches that floor: K re-ordered so all LDS addresses are one
// base register + immediate offset, no cndmask masking (garbage M-rows/N-cols
// fall into unstored D elements; the lone K-pad element uses an LDS zero word).

typedef __attribute__((ext_vector_type(2))) float v2f;
typedef __attribute__((ext_vector_type(8))) float v8f;

namespace {
constexpr int kH = 48, kW = 24;
// LDS region offsets (floats), one flat array:
constexpr int SXo = 0;             // padded X tile: 52 rows * pitch 29
constexpr int SYo = 1508;          // padded Y tile: 56 rows * pitch 29
constexpr int STX = 3132;          // X stats, interleaved (mu, rs): 48*24*2
constexpr int STY = 5436;          // Y stats, interleaved (mu, rs): 52*24*2
constexpr int ZW  = 7932;          // zero word (K-padding reads)
constexpr int LDSN = 7936;
}

#define WMMA_F32(Af, Bf, Cf) \
  __builtin_amdgcn_wmma_f32_16x16x4_f32(false, (Af), false, (Bf), (short)0, (Cf), false, false)

__global__ __launch_bounds__(256)
void ncc_wmma_kernel(const float* __restrict__ X, const float* __restrict__ Y,
                     float* __restrict__ out) {
  __shared__ float lds[LDSN];

  const int tid = threadIdx.x;
  const int blk = blockIdx.x;                 // = b*D + d
  const float* Xs = X + (size_t)blk * (kH * kW);
  const float* Ys = Y + (size_t)blk * (kH * kW);

  if (tid == 0) { lds[ZW] = 0.f; lds[ZW + 1] = 0.f; lds[ZW + 2] = 0.f; lds[ZW + 3] = 0.f; }

  // ---- stage zero-padded tiles into LDS (cols 0..27 used, col 28 pad) ----
  for (int i = tid; i < 52 * 28; i += 256) {
    int r = i / 28, c = i % 28;
    int gr = r - 2, gc = c - 2;
    float v = (gr >= 0 && gr < kH && gc >= 0 && gc < kW) ? Xs[gr * kW + gc] : 0.f;
    lds[SXo + r * 29 + c] = v;
  }
  for (int i = tid; i < 56 * 28; i += 256) {
    int r = i / 28, c = i % 28;
    int gr = r - 4, gc = c - 2;
    float v = (gr >= 0 && gr < kH && gc >= 0 && gc < kW) ? Ys[gr * kW + gc] : 0.f;
    lds[SYo + r * 29 + c] = v;
  }
  __syncthreads();

  // ---- per-patch (mu, 1/(std+0.01)) interleaved for ds_load_b64 fetches ----
  for (int i = tid; i < kH * kW; i += 256) {
    int x = i / kW, y = i % kW;
    float s = 0.f, ss = 0.f;
#pragma unroll
    for (int u = 0; u < 5; ++u)
#pragma unroll
      for (int v = 0; v < 5; ++v) {
        float t = lds[SXo + (x + u) * 29 + y + v];
        s += t; ss += t * t;
      }
    float mu  = s * (1.f / 25.f);
    float var = ss * (1.f / 25.f) - mu * mu;
    var = var > 0.f ? var : 0.f;
    lds[STX + 2 * i]     = mu;
    lds[STX + 2 * i + 1] = 1.f / (sqrtf(var) + 0.01f);
  }
  for (int i = tid; i < 52 * kW; i += 256) {
    int x = i / kW, y = i % kW;
    float s = 0.f, ss = 0.f;
#pragma unroll
    for (int u = 0; u < 5; ++u)
#pragma unroll
      for (int v = 0; v < 5; ++v) {
        float t = lds[SYo + (x + u) * 29 + y + v];
        s += t; ss += t * t;
      }
    float mu  = s * (1.f / 25.f);
    float var = ss * (1.f / 25.f) - mu * mu;
    var = var > 0.f ? var : 0.f;
    lds[STY + 2 * i]     = mu;
    lds[STY + 2 * i + 1] = 1.f / (sqrtf(var) + 0.01f);
  }
  __syncthreads();

  // ---- per-lane setup.  K mapping (A and B identical):
  //   steps 0..4: k=(u=s, v=slot)  slot = (hi?2:0)+component
  //   step  5   : k=(u=slot, v=4)
  //   step  6   : slot0(lo) = (4,4); all other slots zero (LDS zero word)
  const int  lane = tid & 31;
  const int  l16  = lane & 15;
  const bool hi   = lane >= 16;
  const int  mt   = tid >> 5;                 // wave id = M tile 0..7
  const int  ac   = mt * 16 + l16;            // M row (= a*24 + c)
  const bool acv  = ac < 120;
  const int  aa   = acv ? ac / 24 : 0;        // clamped; garbage rows unstored
  const int  cc   = acv ? ac % 24 : 0;
  const int  h2   = hi ? 2 : 0;
  const int  h58  = hi ? 58 : 0;
  const int  yB1  = 16 + l16;
  const bool y1v  = yB1 < kW;
  const int  yc1  = y1v ? yB1 : 0;            // clamped; garbage cols unstored

  const int baseA = SYo + aa * 29 + cc;       // Y patches (A matrix, row ac)
  const int RA    = baseA + h2;
  const int R5A   = baseA + 4 + h58;
  int       R6A   = hi ? ZW : baseA + 120;

  const int baseB0 = SXo + l16;               // X patches (B matrix, col y)
  const int RB0    = baseB0 + h2;
  const int R5B0   = baseB0 + 4 + h58;
  int       R6B0   = hi ? ZW : baseB0 + 120;

  const int baseB1 = SXo + yc1;
  const int RB1    = baseB1 + h2;
  const int R5B1   = baseB1 + 4 + h58;
  int       R6B1   = hi ? ZW : baseB1 + 120;
  const int inc6   = hi ? 0 : 29;             // zero word is static for hi lanes

  const int iSX0 = STX + l16 * 2;             // + x*48
  const int iSX1 = STX + yc1 * 2;
  const int  rowb = mt * 16 + (hi ? 8 : 0);   // D rows rowb..rowb+7
  const bool rowv = rowb < 120;               // single loop-invariant predicate
  const int  rowc = rowv ? rowb : 0;
  const int  iSY  = STY + rowc * 2;           // + x*48 + 2*i (imm)

  // running global pointer; per-i / acc1 offsets become 24-bit store immediates
  float* P = out + ((size_t)blk * 120 + rowc) * (size_t)(kH * kW) + l16;

  int x29 = 0, x48 = 0;
  for (int x = 0; x < kH; ++x) {
    v8f acc0 = {};   // D cols y = 0..15
    v8f acc1 = {};   // D cols y = 16..23 (pad to 31, unstored)
#pragma unroll
    for (int s = 0; s < 5; ++s) {             // k = (u=s, v=slot)
      v2f a, b0, b1;
      a.x  = lds[RA  + x29 + s * 29];  a.y  = lds[RA  + x29 + s * 29 + 1];
      b0.x = lds[RB0 + x29 + s * 29];  b0.y = lds[RB0 + x29 + s * 29 + 1];
      b1.x = lds[RB1 + x29 + s * 29];  b1.y = lds[RB1 + x29 + s * 29 + 1];
      acc0 = WMMA_F32(a, b0, acc0);
      acc1 = WMMA_F32(a, b1, acc1);
    }
    {                                         // step 5: k = (u=slot, v=4)
      v2f a, b0, b1;
      a.x  = lds[R5A  + x29];  a.y  = lds[R5A  + x29 + 29];
      b0.x = lds[R5B0 + x29];  b0.y = lds[R5B0 + x29 + 29];
      b1.x = lds[R5B1 + x29];  b1.y = lds[R5B1 + x29 + 29];
      acc0 = WMMA_F32(a, b0, acc0);
      acc1 = WMMA_F32(a, b1, acc1);
    }
    {                                         // step 6: lone (4,4) element
      v2f a, b0, b1;
      a.x  = lds[R6A];   a.y  = 0.f;
      b0.x = lds[R6B0];  b0.y = 0.f;
      b1.x = lds[R6B1];  b1.y = 0.f;
      acc0 = WMMA_F32(a, b0, acc0);
      acc1 = WMMA_F32(a, b1, acc1);
    }

    // ---- epilogue: NCC fixup + store (two 64B runs per store instruction) ----
    float mx0 = lds[iSX0 + x48], rx0 = lds[iSX0 + x48 + 1];
    float mx1 = lds[iSX1 + x48], rx1 = lds[iSX1 + x48 + 1];
    if (rowv) {
#pragma unroll
      for (int i = 0; i < 8; ++i) {
        float my = lds[iSY + x48 + 2 * i];
        float ry = lds[iSY + x48 + 2 * i + 1];
        P[i * (kH * kW)] = (acc0[i] * (1.f / 25.f) - mx0 * my) * (rx0 * ry);
        if (y1v)
          P[i * (kH * kW) + 16] = (acc1[i] * (1.f / 25.f) - mx1 * my) * (rx1 * ry);
      }
    }
    P += kW;
    x29 += 29;
    x48 += 48;
    R6A += inc6; R6B0 += inc6; R6B1 += inc6;
  }
}

extern "C" void kernel_launch(void* const* d_in, const int* in_sizes, int n_in,
                              void* d_out, int out_size, void* d_ws, size_t ws_size,
                              hipStream_t stream) {
  const float* X = (const float*)d_in[0];
  const float* Y = (const float*)d_in[1];
  float* out = (float*)d_out;
  const int nslices = in_sizes[0] / (kH * kW);   // B*D = 1024
  hipLaunchKernelGGL(ncc_wmma_kernel, dim3(nslices), dim3(256), 0, stream,
                     X, Y, out);
}